// PostPatchAdaptiveGraphLearner_5909875000330
// MI455X (gfx1250) — compile-verified
//
#include <hip/hip_runtime.h>
#include <math.h>

// Problem dims (compile-time constants, match reference)
#define B_   16
#define N_   1024
#define P_   72
#define D_   96
#define ND_  64
#define H_   4
#define K_   16

typedef __attribute__((ext_vector_type(2))) float v2f;
typedef __attribute__((ext_vector_type(8))) float v8f;

__device__ __forceinline__ float wave_max(float v) {
#pragma unroll
    for (int off = 16; off > 0; off >>= 1) v = fmaxf(v, __shfl_xor(v, off, 32));
    return v;
}
__device__ __forceinline__ float wave_sum(float v) {
#pragma unroll
    for (int off = 16; off > 0; off >>= 1) v += __shfl_xor(v, off, 32);
    return v;
}

// ---------------------------------------------------------------------------
// Kernel 1: mean over P of patch_features (B,N,P,D) -> node_repr (B,N,D)
// One block per (b,n); 96 threads, coalesced over D each p-step.
// ---------------------------------------------------------------------------
__global__ void meanpool_kernel(const float* __restrict__ pf, float* __restrict__ node) {
    const int bn = blockIdx.x;
    const int d  = threadIdx.x;              // 0..95
    const float* p = pf + (size_t)bn * P_ * D_ + d;
    float s = 0.f;
#pragma unroll 8
    for (int i = 0; i < P_; ++i) s += p[(size_t)i * D_];
    node[(size_t)bn * D_ + d] = s * (1.0f / P_);
}

// ---------------------------------------------------------------------------
// Kernel 2: dyn_emb = relu(LN(node@dW1+db1)) @ dW2 + db2   per row (64 thr)
// ---------------------------------------------------------------------------
__global__ void dynemb_kernel(const float* __restrict__ node,
                              const float* __restrict__ dW1, const float* __restrict__ db1,
                              const float* __restrict__ lng, const float* __restrict__ lnb,
                              const float* __restrict__ dW2, const float* __restrict__ db2,
                              float* __restrict__ dyn) {
    __shared__ float xs[D_];
    __shared__ float h1s[ND_];
    __shared__ float red[ND_];
    const int bn = blockIdx.x;
    const int t  = threadIdx.x;              // 0..63
    for (int i = t; i < D_; i += ND_) xs[i] = node[(size_t)bn * D_ + i];
    __syncthreads();
    float acc = db1[t];
#pragma unroll 8
    for (int i = 0; i < D_; ++i) acc += xs[i] * dW1[i * ND_ + t];
    // LayerNorm over ND_=64
    red[t] = acc; __syncthreads();
    for (int s = 32; s > 0; s >>= 1) { if (t < s) red[t] += red[t + s]; __syncthreads(); }
    const float mu = red[0] * (1.0f / ND_);
    __syncthreads();
    const float dif = acc - mu;
    red[t] = dif * dif; __syncthreads();
    for (int s = 32; s > 0; s >>= 1) { if (t < s) red[t] += red[t + s]; __syncthreads(); }
    const float var = red[0] * (1.0f / ND_);
    __syncthreads();
    float h = dif * rsqrtf(var + 1e-5f) * lng[t] + lnb[t];
    h = fmaxf(h, 0.0f);
    h1s[t] = h; __syncthreads();
    float o = db2[t];
#pragma unroll 8
    for (int i = 0; i < ND_; ++i) o += h1s[i] * dW2[i * ND_ + t];
    dyn[(size_t)bn * ND_ + t] = o;
}

// ---------------------------------------------------------------------------
// Kernel 3: fused  (A @ E2[h]) -> relu -> /T -> softmax -> top-16 -> renorm
// Block = 256 threads (8 wave32). Each block owns a 16-row x 1024-col strip
// held in 64KB LDS. GEMM uses V_WMMA_F32_16X16X4_F32, K=ND_=64 -> 16 k-steps.
//   static : grid = H*64,     A = E1 + h*N*ND,   out batch = h   (= bid>>6)
//   dynamic: grid = B*H*64,   A = dyn + b*N*ND,  out batch = b*H+h (= bid>>6)
// ---------------------------------------------------------------------------
__global__ __launch_bounds__(256)
void adj_fused_kernel(const float* __restrict__ Abase, const float* __restrict__ E2,
                      const float* __restrict__ temperature,
                      float* __restrict__ out, int dynamicMode) {
    __shared__ float sc[16][N_];            // 64 KB score strip
    const int bid     = blockIdx.x;
    const int rowTile = bid & 63;
    int h, aIdx;
    if (dynamicMode) {
        const int bh = bid >> 6;            // b*H + h
        h    = bh & (H_ - 1);
        aIdx = bh >> 2;                     // b
    } else {
        h    = bid >> 6;
        aIdx = h;
    }
    const float* A  = Abase + (size_t)aIdx * N_ * ND_;
    const float* Bm = E2    + (size_t)h    * ND_ * N_;
    float* obase    = out   + (size_t)(bid >> 6) * N_ * N_;

    const int r0   = rowTile * 16;
    const int lane = threadIdx.x & 31;
    const int w    = threadIdx.x >> 5;      // 8 waves; wave owns cols [w*128, w*128+128)
    const int hl   = lane >> 4;             // lane-half selector
    const int ml   = lane & 15;
    const float invT = 1.0f / temperature[h];

    v8f acc[8] = {};
    const float* Arow = A + (size_t)(r0 + ml) * ND_;
    const int nbase = w * 128 + ml;
#pragma unroll
    for (int ks = 0; ks < 16; ++ks) {
        const int k0 = ks * 4 + 2 * hl;     // ISA A-layout: vgpr0 K=k0, vgpr1 K=k0+1
        v2f a; a.x = Arow[k0]; a.y = Arow[k0 + 1];
#pragma unroll
        for (int t = 0; t < 8; ++t) {
            const float* bp = Bm + (size_t)k0 * N_ + nbase + t * 16;
            v2f b; b.x = bp[0]; b.y = bp[N_];
            acc[t] = __builtin_amdgcn_wmma_f32_16x16x4_f32(
                false, a, false, b, (short)0, acc[t], false, false);
        }
    }
    // spill strip to LDS with relu and 1/T (ISA C-layout: vgpr i -> M = i + 8*hl)
#pragma unroll
    for (int t = 0; t < 8; ++t) {
        const int n = w * 128 + t * 16 + ml;
#pragma unroll
        for (int i = 0; i < 8; ++i) {
            sc[i + 8 * hl][n] = fmaxf(acc[t][i], 0.0f) * invT;
        }
    }
    __syncthreads();

    // Each wave: softmax + top-16 + renormalize for rows 2w, 2w+1
    for (int rr = 0; rr < 2; ++rr) {
        const int r  = 2 * w + rr;
        float* row   = sc[r];
        float rmax = -3.4e38f;
        for (int c = lane; c < N_; c += 32) rmax = fmaxf(rmax, row[c]);
        rmax = wave_max(rmax);
        float dsum = 0.f;
        for (int c = lane; c < N_; c += 32) dsum += __expf(row[c] - rmax);
        dsum = wave_sum(dsum);

        float topv[K_]; int topi[K_];
        float spsum = 0.f;
#pragma unroll
        for (int j = 0; j < K_; ++j) {
            float bv = -3.4e38f; int bi = 0x7fffffff;
            for (int c = lane; c < N_; c += 32) {
                const float v = row[c];
                if (v > bv) { bv = v; bi = c; }
            }
#pragma unroll
            for (int off = 16; off > 0; off >>= 1) {
                const float ov = __shfl_xor(bv, off, 32);
                const int   oi = __shfl_xor(bi, off, 32);
                if (ov > bv || (ov == bv && oi < bi)) { bv = ov; bi = oi; }
            }
            topv[j] = bv; topi[j] = bi;
            spsum += __expf(bv - rmax);
            row[bi] = -3.4e38f;                         // knock out selected entry
            asm volatile("s_wait_dscnt 0" ::: "memory"); // LDS RAW fence (in-wave)
        }
        const float invd  = 1.0f / dsum;
        const float denom = spsum * invd + 1e-8f;       // sp.sum + ROW_EPS
        float pv[K_];
#pragma unroll
        for (int j = 0; j < K_; ++j) pv[j] = (__expf(topv[j] - rmax) * invd) / denom;

        float* og = obase + (size_t)(r0 + r) * N_;
        for (int c = lane; c < N_; c += 32) {
            float v = 0.f;
#pragma unroll
            for (int j = 0; j < K_; ++j) if (c == topi[j]) v = pv[j];
            og[c] = v;
        }
    }
}

// ---------------------------------------------------------------------------
// Kernel 4: fuse static+dynamic, edge MLP (4 -> 8 -> 1), sigmoid * mean_h
// One element of (B,N,N) per lane.
// ---------------------------------------------------------------------------
__global__ void fuse_kernel(const float* __restrict__ statA, const float* __restrict__ dynA,
                            const float* __restrict__ fwp,
                            const float* __restrict__ eW1, const float* __restrict__ eb1,
                            const float* __restrict__ eW2, const float* __restrict__ eb2,
                            float* __restrict__ fin) {
    const size_t idx = (size_t)blockIdx.x * blockDim.x + threadIdx.x;
    const size_t NN  = (size_t)N_ * N_;
    const size_t b   = idx / NN;
    const size_t rem = idx - b * NN;
    const float fw = fwp[0];
    float f[H_];
    float mean = 0.f;
#pragma unroll
    for (int h = 0; h < H_; ++h) {
        const float s = statA[(size_t)h * NN + rem];
        const float d = dynA[(b * H_ + h) * NN + rem];
        f[h] = (1.0f - fw) * s + fw * d;
        mean += f[h];
    }
    mean *= (1.0f / H_);
    float ew = eb2[0];
#pragma unroll
    for (int j = 0; j < 2 * H_; ++j) {
        float hj = eb1[j];
#pragma unroll
        for (int h = 0; h < H_; ++h) hj += f[h] * eW1[h * (2 * H_) + j];
        ew += fmaxf(hj, 0.0f) * eW2[j];
    }
    const float sig = 1.0f / (1.0f + __expf(-ew));
    fin[idx] = sig * mean;
}

// ---------------------------------------------------------------------------
extern "C" void kernel_launch(void* const* d_in, const int* in_sizes, int n_in,
                              void* d_out, int out_size, void* d_ws, size_t ws_size,
                              hipStream_t stream) {
    (void)in_sizes; (void)n_in; (void)out_size; (void)ws_size;
    const float* pf   = (const float*)d_in[0];   // (B,N,P,D)
    const float* E1   = (const float*)d_in[1];   // (H,N,ND)
    const float* E2   = (const float*)d_in[2];   // (H,ND,N)
    const float* temp = (const float*)d_in[3];   // (H,)
    const float* fw   = (const float*)d_in[4];   // scalar
    const float* dW1  = (const float*)d_in[5];
    const float* db1  = (const float*)d_in[6];
    const float* lng  = (const float*)d_in[7];
    const float* lnb  = (const float*)d_in[8];
    const float* dW2  = (const float*)d_in[9];
    const float* db2  = (const float*)d_in[10];
    const float* eW1  = (const float*)d_in[11];
    const float* eb1  = (const float*)d_in[12];
    const float* eW2  = (const float*)d_in[13];
    const float* eb2  = (const float*)d_in[14];

    float* out   = (float*)d_out;                        // concat: final, static, dynamic
    float* fin   = out;                                  // (B,N,N)
    float* statA = out + (size_t)B_ * N_ * N_;           // (H,N,N)
    float* dynA  = statA + (size_t)H_ * N_ * N_;         // (B,H,N,N)

    float* node = (float*)d_ws;                          // (B,N,D)
    float* dyn  = node + (size_t)B_ * N_ * D_;           // (B,N,ND)

    meanpool_kernel<<<B_ * N_, D_, 0, stream>>>(pf, node);
    dynemb_kernel<<<B_ * N_, ND_, 0, stream>>>(node, dW1, db1, lng, lnb, dW2, db2, dyn);
    adj_fused_kernel<<<H_ * (N_ / 16), 256, 0, stream>>>(E1, E2, temp, statA, 0);
    adj_fused_kernel<<<B_ * H_ * (N_ / 16), 256, 0, stream>>>(dyn, E2, temp, dynA, 1);
    fuse_kernel<<<(B_ * N_ * N_) / 256, 256, 0, stream>>>(statA, dynA, fw, eW1, eb1, eW2, eb2, fin);
}